// LatentDynamicsModel_15135464751511
// MI455X (gfx1250) — compile-verified
//
#include <hip/hip_runtime.h>

typedef __attribute__((ext_vector_type(2))) float v2f;
typedef __attribute__((ext_vector_type(8))) float v8f;

#define WMMA_F32(a, b, c) \
  __builtin_amdgcn_wmma_f32_16x16x4_f32(false, (a), false, (b), (short)0, (c), false, false)

// ---------------------------------------------------------------- helpers
__device__ __forceinline__ float block_sum256(float v, float* red) {
  #pragma unroll
  for (int m = 16; m >= 1; m >>= 1) v += __shfl_xor(v, m, 32);
  int w = threadIdx.x >> 5;
  if ((threadIdx.x & 31) == 0) red[w] = v;
  __syncthreads();
  float r = 0.f;
  if (threadIdx.x < 8) r = red[threadIdx.x];
  if (w == 0) {
    #pragma unroll
    for (int m = 4; m >= 1; m >>= 1) r += __shfl_xor(r, m, 32);
  }
  __syncthreads();
  return r;  // valid on thread 0
}

// ---------------------------------------------------------------- stage 0
__global__ void zero_k(float* stats) { stats[threadIdx.x] = 0.f; }

// ---------------------------------------------------------------- conv1: (64,3,9,64,64) -> (64,32,9,32,32)
__global__ void __launch_bounds__(256) conv1_k(const float* __restrict__ x,
                                               const float* __restrict__ w,
                                               const float* __restrict__ bias,
                                               float* __restrict__ out,
                                               float* __restrict__ stats) {
  __shared__ float wsm[144];
  __shared__ float red[8];
  int bid = blockIdx.x;                    // ((b*32+co)*9+d)*4 + q
  int q = bid & 3;
  int t2 = bid >> 2;
  int d = t2 % 9;
  int t3 = t2 / 9;
  int co = t3 & 31;
  int b = t3 >> 5;
  int t = threadIdx.x;
  if (t < 144) wsm[t] = w[co * 144 + t];
  __syncthreads();
  int p = q * 256 + t;                     // 0..1023 over 32x32
  int y = p >> 5, xo = p & 31;
  float acc = bias[co];
  for (int ci = 0; ci < 3; ++ci)
    for (int kd = 0; kd < 3; ++kd) {
      int id = d + kd - 1;
      if (id < 0 || id > 8) continue;
      const float* src = x + ((((b * 3 + ci) * 9 + id)) << 12);
      for (int kh = 0; kh < 4; ++kh) {
        int iy = y * 2 + kh - 1;
        if (iy < 0 || iy > 63) continue;
        for (int kw = 0; kw < 4; ++kw) {
          int ix = xo * 2 + kw - 1;
          if (ix < 0 || ix > 63) continue;
          acc += wsm[((ci * 3 + kd) * 4 + kh) * 4 + kw] * src[(iy << 6) + ix];
        }
      }
    }
  out[(((b * 32 + co) * 9 + d) << 10) + p] = acc;
  float ts = block_sum256(acc, red);
  float ts2 = block_sum256(acc * acc, red);
  if (t == 0) { atomicAdd(&stats[co], ts); atomicAdd(&stats[32 + co], ts2); }
}

// ---------------------------------------------------------------- BN+ReLU in-place
__global__ void __launch_bounds__(256) bn_relu_k(float* __restrict__ buf,
                                                 const float* __restrict__ g,
                                                 const float* __restrict__ bb,
                                                 const float* __restrict__ sum,
                                                 const float* __restrict__ sq,
                                                 int C, int sd, int total, float invN) {
  int i = blockIdx.x * 256 + threadIdx.x;
  if (i >= total) return;
  int c = (i / sd) % C;
  float mean = sum[c] * invN;
  float var = sq[c] * invN - mean * mean;
  float v = buf[i];
  v = (v - mean) * rsqrtf(var + 1e-5f) * g[c] + bb[c];
  buf[i] = v > 0.f ? v : 0.f;
}

// ---------------------------------------------------------------- conv2: (64,32,9,32,32) -> (64,64,9,16,16)
__global__ void __launch_bounds__(256) conv2_k(const float* __restrict__ h1,
                                               const float* __restrict__ w,
                                               const float* __restrict__ bias,
                                               float* __restrict__ out,
                                               float* __restrict__ stats) {
  __shared__ float wsm[1536];
  __shared__ float red[8];
  int bid = blockIdx.x;                    // (b*64+co)*9 + d
  int d = bid % 9;
  int t3 = bid / 9;
  int co = t3 & 63;
  int b = t3 >> 6;
  int t = threadIdx.x;
  for (int i = t; i < 1536; i += 256) wsm[i] = w[co * 1536 + i];
  __syncthreads();
  int y = t >> 4, xo = t & 15;
  float acc = bias[co];
  for (int ci = 0; ci < 32; ++ci) {
    const float* src = h1 + (((b * 32 + ci) * 9) << 10);
    const float* wp = wsm + ci * 48;
    for (int kd = 0; kd < 3; ++kd) {
      int id = d + kd - 1;
      if (id < 0 || id > 8) continue;
      const float* s2 = src + (id << 10);
      for (int kh = 0; kh < 4; ++kh) {
        int iy = y * 2 + kh - 1;
        if (iy < 0 || iy > 31) continue;
        for (int kw = 0; kw < 4; ++kw) {
          int ix = xo * 2 + kw - 1;
          if (ix < 0 || ix > 31) continue;
          acc += wp[(kd * 4 + kh) * 4 + kw] * s2[(iy << 5) + ix];
        }
      }
    }
  }
  out[(((b * 64 + co) * 9 + d) << 8) + t] = acc;
  float ts = block_sum256(acc, red);
  float ts2 = block_sum256(acc * acc, red);
  if (t == 0) { atomicAdd(&stats[64 + co], ts); atomicAdd(&stats[128 + co], ts2); }
}

// ---------------------------------------------------------------- f_t = proj(h2)[:, :, -1], layout ft[m][f], m=b*256+y*16+x
__global__ void __launch_bounds__(256) ft_k(const float* __restrict__ h2,
                                            const float* __restrict__ pw,
                                            const float* __restrict__ pb,
                                            float* __restrict__ ft) {
  int gid = blockIdx.x * 256 + threadIdx.x;   // 16384*128
  int f = gid & 127, m = gid >> 7;
  int b = m >> 8, p = m & 255;
  float acc = pb[f];
  const float* base = h2 + ((b * 64) * 9 + 8) * 256 + p;   // c stride = 9*256
  for (int c = 0; c < 64; ++c) acc += pw[f * 64 + c] * base[c * 2304];
  ft[gid] = acc;
}

// ---------------------------------------------------------------- Bh[h][f][g] = embed_w[g,f]*queries[h,f]; cb2[k] = |codebook_k|^2
__global__ void __launch_bounds__(256) prep_k(const float* __restrict__ ew,
                                              const float* __restrict__ qr,
                                              const float* __restrict__ cb,
                                              float* __restrict__ Bh,
                                              float* __restrict__ cb2) {
  int t = blockIdx.x * 256 + threadIdx.x;
  if (t < 8 * 128 * 128) {
    int h = t >> 14, f = (t >> 7) & 127, g = t & 127;
    Bh[t] = ew[g * 128 + f] * qr[h * 128 + f];
  } else if (t < 8 * 128 * 128 + 1024) {
    int k = t - 8 * 128 * 128;
    float s = 0.f;
    for (int f = 0; f < 128; ++f) { float v = cb[k * 128 + f]; s += v * v; }
    cb2[k] = s;
  }
}

// ---------------------------------------------------------------- z_e GEMM: for each h, ze = ft(16384x128) * Bh(128x128) + embed_b
__global__ void __launch_bounds__(256) ze_gemm_k(const float* __restrict__ A,
                                                 const float* __restrict__ Bh,
                                                 const float* __restrict__ eb,
                                                 float* __restrict__ ze) {
  int wave = blockIdx.x * 8 + (threadIdx.x >> 5);   // 16384 waves
  int lane = threadIdx.x & 31;
  int half = lane >> 4, l16 = lane & 15;
  int h = wave >> 11;
  int rem = wave & 2047;
  int mt = rem >> 1, nt = rem & 1;
  int mrow = mt << 4;
  int nbase = nt << 6;
  const float* Bp = Bh + (h << 14);
  const float* Arow = A + (mrow + l16) * 128;
  v8f c0 = {}, c1 = {}, c2 = {}, c3 = {};
  for (int k0 = 0; k0 < 128; k0 += 4) {
    v2f a;
    a.x = Arow[k0 + 2 * half];
    a.y = Arow[k0 + 2 * half + 1];
    const float* bp = Bp + (k0 + 2 * half) * 128 + nbase + l16;
    v2f b0, b1, b2, b3;
    b0.x = bp[0];  b0.y = bp[128];
    b1.x = bp[16]; b1.y = bp[144];
    b2.x = bp[32]; b2.y = bp[160];
    b3.x = bp[48]; b3.y = bp[176];
    c0 = WMMA_F32(a, b0, c0);
    c1 = WMMA_F32(a, b1, c1);
    c2 = WMMA_F32(a, b2, c2);
    c3 = WMMA_F32(a, b3, c3);
  }
  #pragma unroll
  for (int r = 0; r < 8; ++r) {
    int gr = (h << 14) + mrow + r + (half << 3);
    float* zp = ze + gr * 128 + nbase + l16;
    zp[0]  = c0[r] + eb[nbase + l16];
    zp[16] = c1[r] + eb[nbase + l16 + 16];
    zp[32] = c2[r] + eb[nbase + l16 + 32];
    zp[48] = c3[r] + eb[nbase + l16 + 48];
  }
}

// ---------------------------------------------------------------- dist GEMM + argmin: 131072 rows x 1024 codes, K=128
// A panel (16x128 per wave) hoisted into registers once, reused for all 16 N tiles.
__global__ void __launch_bounds__(256) dist_argmin_k(const float* __restrict__ ze,
                                                     const float* __restrict__ cb,
                                                     const float* __restrict__ cb2,
                                                     int* __restrict__ idx) {
  int wave = blockIdx.x * 8 + (threadIdx.x >> 5);   // 8192 waves, 16 rows each
  int lane = threadIdx.x & 31;
  int half = lane >> 4, l16 = lane & 15;
  int mrow = wave << 4;
  const float* Arow = ze + (mrow + l16) * 128;
  v2f areg[32];
  #pragma unroll
  for (int i = 0; i < 32; ++i) {
    areg[i].x = Arow[4 * i + 2 * half];
    areg[i].y = Arow[4 * i + 2 * half + 1];
  }
  float bestS[8];
  int bestI[8];
  #pragma unroll
  for (int r = 0; r < 8; ++r) { bestS[r] = 3.4e38f; bestI[r] = 0; }
  for (int nt = 0; nt < 16; ++nt) {
    int nbase = nt << 6;
    v8f c0 = {}, c1 = {}, c2 = {}, c3 = {};
    #pragma unroll
    for (int ks = 0; ks < 32; ++ks) {
      int kk = 4 * ks + 2 * half;
      const float* cbp = cb + (nbase + l16) * 128 + kk;   // B[f][n] = cb[n*128+f]
      v2f b0, b1, b2, b3;
      b0.x = cbp[0];    b0.y = cbp[1];
      b1.x = cbp[2048]; b1.y = cbp[2049];
      b2.x = cbp[4096]; b2.y = cbp[4097];
      b3.x = cbp[6144]; b3.y = cbp[6145];
      c0 = WMMA_F32(areg[ks], b0, c0);
      c1 = WMMA_F32(areg[ks], b1, c1);
      c2 = WMMA_F32(areg[ks], b2, c2);
      c3 = WMMA_F32(areg[ks], b3, c3);
    }
    #pragma unroll
    for (int s = 0; s < 4; ++s) {
      int n = nbase + s * 16 + l16;
      float cc2 = cb2[n];
      const v8f& c = (s == 0) ? c0 : (s == 1) ? c1 : (s == 2) ? c2 : c3;
      #pragma unroll
      for (int r = 0; r < 8; ++r) {
        float sc = cc2 - 2.0f * c[r];
        if (sc < bestS[r] || (sc == bestS[r] && n < bestI[r])) { bestS[r] = sc; bestI[r] = n; }
      }
    }
  }
  // reduce across the 16 lanes of each half (masks 1..8 never cross bit 4)
  #pragma unroll
  for (int m = 1; m <= 8; m <<= 1) {
    #pragma unroll
    for (int r = 0; r < 8; ++r) {
      float os = __shfl_xor(bestS[r], m, 32);
      int oi = __shfl_xor(bestI[r], m, 32);
      if (os < bestS[r] || (os == bestS[r] && oi < bestI[r])) { bestS[r] = os; bestI[r] = oi; }
    }
  }
  #pragma unroll
  for (int r = 0; r < 8; ++r)
    if (l16 == r) idx[mrow + (half << 3) + r] = bestI[r];
}

// ---------------------------------------------------------------- scatter z_q_st to output + loss accumulation
__global__ void __launch_bounds__(256) out_k(const float* __restrict__ ze,
                                             const float* __restrict__ cb,
                                             const int* __restrict__ idx,
                                             float* __restrict__ out,
                                             float* __restrict__ lossAcc) {
  __shared__ float red[8];
  int gid = blockIdx.x * 256 + threadIdx.x;   // 16,777,216
  int g = gid & 127;
  int row = gid >> 7;
  int h = row >> 14;
  int m = row & 16383;
  int b = m >> 8, p = m & 255;
  int k = idx[row];
  float zq = cb[k * 128 + g];
  float zev = ze[(size_t)row * 128 + g];
  float d = zq - zev;
  out[(size_t)(((b << 3) + h) * 128 + g) * 256 + p] = zev + d;   // z_q_st forward == z_q
  float ts = block_sum256(d * d, red);
  if (threadIdx.x == 0) atomicAdd(lossAcc, ts);
}

__global__ void fin_k(const float* __restrict__ lossAcc, float* __restrict__ out) {
  out[16777216] = 1.25f * lossAcc[0] / 16777216.0f;
}

// ---------------------------------------------------------------- launch
extern "C" void kernel_launch(void* const* d_in, const int* in_sizes, int n_in,
                              void* d_out, int out_size, void* d_ws, size_t ws_size,
                              hipStream_t stream) {
  const float* x_seq   = (const float*)d_in[0];
  const float* conv1_w = (const float*)d_in[1];
  const float* conv1_b = (const float*)d_in[2];
  const float* bn1_g   = (const float*)d_in[3];
  const float* bn1_b   = (const float*)d_in[4];
  const float* conv2_w = (const float*)d_in[5];
  const float* conv2_b = (const float*)d_in[6];
  const float* bn2_g   = (const float*)d_in[7];
  const float* bn2_b   = (const float*)d_in[8];
  const float* proj_w  = (const float*)d_in[9];
  const float* proj_b  = (const float*)d_in[10];
  const float* queries = (const float*)d_in[11];
  const float* embed_w = (const float*)d_in[12];
  const float* embed_b = (const float*)d_in[13];
  const float* codebook= (const float*)d_in[14];
  float* out = (float*)d_out;

  float* ws = (float*)d_ws;
  float* buf1  = ws;                         // 18,874,368 f : conv1 out / h1 (in-place BN)
  float* buf2  = buf1 + 18874368;            //  9,437,184 f : conv2 out / h2
  float* ft    = buf2 + 9437184;             //  2,097,152 f
  float* ze    = ft + 2097152;               // 16,777,216 f
  float* Bh    = ze + 16777216;              //    131,072 f
  float* cb2   = Bh + 131072;                //      1,024 f
  float* stats = cb2 + 1024;                 //        256 f (sum1,sq1,sum2,sq2,lossAcc@192)
  int*   idxb  = (int*)(stats + 256);        //    131,072 i

  zero_k<<<1, 256, 0, stream>>>(stats);
  conv1_k<<<73728, 256, 0, stream>>>(x_seq, conv1_w, conv1_b, buf1, stats);
  bn_relu_k<<<73728, 256, 0, stream>>>(buf1, bn1_g, bn1_b, stats, stats + 32,
                                       32, 9 * 1024, 18874368, 1.0f / 589824.0f);
  conv2_k<<<36864, 256, 0, stream>>>(buf1, conv2_w, conv2_b, buf2, stats);
  bn_relu_k<<<36864, 256, 0, stream>>>(buf2, bn2_g, bn2_b, stats + 64, stats + 128,
                                       64, 9 * 256, 9437184, 1.0f / 147456.0f);
  ft_k<<<8192, 256, 0, stream>>>(buf2, proj_w, proj_b, ft);
  prep_k<<<517, 256, 0, stream>>>(embed_w, queries, codebook, Bh, cb2);
  ze_gemm_k<<<2048, 256, 0, stream>>>(ft, Bh, embed_b, ze);
  dist_argmin_k<<<1024, 256, 0, stream>>>(ze, codebook, cb2, idxb);
  out_k<<<65536, 256, 0, stream>>>(ze, codebook, idxb, out, stats + 192);
  fin_k<<<1, 1, 0, stream>>>(stats + 192, out);
}